// CML_85383949844517
// MI455X (gfx1250) — compile-verified
//
#include <hip/hip_runtime.h>

typedef float v2f __attribute__((ext_vector_type(2)));
typedef float v8f __attribute__((ext_vector_type(8)));

#define DIM 128
#define KNEG 64

// One wave (32 lanes) per user. 8 waves / block.
// Per tile of 16 negatives:
//   lane = h*16 + m  (m = negative in tile, h = which half of D)
//   lane accumulates acc0 over d in [64h, 64h+32), acc1 over [64h+32, 64h+64)
//   -> A-matrix (16x4 f32): lanes 0-15 hold A[m,0..1] in v0/v1, lanes 16-31 hold A[m,2..3]
//   one V_WMMA_F32_16X16X4_F32 with B = all-ones reduces to the 16 distances.
__global__ __launch_bounds__(256) void cml_dist_kernel(
    const int* __restrict__ pairs,      // (B,2) [user, item]
    const int* __restrict__ pos_art,    // (B)
    const int* __restrict__ pos_alb,    // (B)
    const int* __restrict__ neg_items,  // (B,K)
    const int* __restrict__ neg_art,    // (B,K)
    const int* __restrict__ neg_alb,    // (B,K)
    const float* __restrict__ Wu,
    const float* __restrict__ Wi,
    const float* __restrict__ Wart,
    const float* __restrict__ Walb,
    float* __restrict__ out_pos,        // (B)
    float* __restrict__ out_dist,       // (B,K)
    float* __restrict__ out_min,        // (B)
    int B)
{
    __shared__ float uvec[8][DIM];       // staged user vector per wave
    __shared__ float bounce[8][32 * 8];  // WMMA D-matrix bounce per wave

    const int lane = threadIdx.x & 31;
    const int wave = threadIdx.x >> 5;
    const int b    = blockIdx.x * 8 + wave;
    if (b >= B) return;                  // wave-uniform

    const float third = 1.0f / 3.0f;

    const int uid = pairs[2 * b + 0];
    const int pid = pairs[2 * b + 1];

    // ---- stage user vector in LDS (coalesced, 4 floats/lane) ----
    float4 u4 = *(const float4*)(Wu + (size_t)uid * DIM + lane * 4);
    *(float4*)(&uvec[wave][lane * 4]) = u4;
    __syncthreads();

    // ---- positive distance: each lane owns 4 dims, wave-reduce ----
    {
        const float* wi_p = Wi   + (size_t)pid        * DIM;
        const float* wa_p = Wart + (size_t)pos_art[b] * DIM;
        const float* wl_p = Walb + (size_t)pos_alb[b] * DIM;
        float4 a = *(const float4*)(wi_p + lane * 4);
        float4 r = *(const float4*)(wa_p + lane * 4);
        float4 l = *(const float4*)(wl_p + lane * 4);
        float accp = 0.0f, s;
        s = u4.x - (a.x + r.x + l.x) * third; accp = fmaf(s, s, accp);
        s = u4.y - (a.y + r.y + l.y) * third; accp = fmaf(s, s, accp);
        s = u4.z - (a.z + r.z + l.z) * third; accp = fmaf(s, s, accp);
        s = u4.w - (a.w + r.w + l.w) * third; accp = fmaf(s, s, accp);
        #pragma unroll
        for (int mask = 16; mask; mask >>= 1)
            accp += __shfl_xor(accp, mask, 32);
        if (lane == 0) out_pos[b] = accp;
    }

    // ---- negative distances ----
    const int h = lane >> 4;             // 0 or 1: which half of D
    const int m = lane & 15;             // negative index within tile
    const int dbase = 64 * h;            // this lane's contiguous 64-float slice
    const float* uw = &uvec[wave][0];

    v2f ones; ones[0] = 1.0f; ones[1] = 1.0f;   // B matrix: all ones (layout-invariant)
    float minv = INFINITY;

    for (int t = 0; t < 4; ++t) {        // 4 tiles x 16 negatives = K
        const int k = t * 16 + m;
        const size_t ri = (size_t)neg_items[(size_t)b * KNEG + k] * DIM;
        const size_t ra = (size_t)neg_art [(size_t)b * KNEG + k] * DIM;
        const size_t rl = (size_t)neg_alb [(size_t)b * KNEG + k] * DIM;

        float acc0 = 0.0f, acc1 = 0.0f;
        #pragma unroll
        for (int j = 0; j < 16; ++j) {   // 16 x float4 = 64 dims for this lane
            const int d0 = dbase + 4 * j;
            float4 wi4 = *(const float4*)(Wi   + ri + d0);
            float4 wa4 = *(const float4*)(Wart + ra + d0);
            float4 wl4 = *(const float4*)(Walb + rl + d0);
            float4 uu  = *(const float4*)(uw + d0);   // LDS broadcast read
            float acc = 0.0f, s;
            s = uu.x - (wi4.x + wa4.x + wl4.x) * third; acc = fmaf(s, s, acc);
            s = uu.y - (wi4.y + wa4.y + wl4.y) * third; acc = fmaf(s, s, acc);
            s = uu.z - (wi4.z + wa4.z + wl4.z) * third; acc = fmaf(s, s, acc);
            s = uu.w - (wi4.w + wa4.w + wl4.w) * third; acc = fmaf(s, s, acc);
            if (j < 8) acc0 += acc; else acc1 += acc;
        }

        // A[m, 2h+0]=acc0, A[m, 2h+1]=acc1 ; D[m,n] = sum_k A[m,k] = distance of neg m
        v2f A; A[0] = acc0; A[1] = acc1;
        v8f c = {};
        c = __builtin_amdgcn_wmma_f32_16x16x4_f32(
                /*neg_a=*/false, A, /*neg_b=*/false, ones,
                /*c_mod=*/(short)0, c, /*reuse_a=*/false, /*reuse_b=*/false);

        // Linearize WMMA D layout through LDS: (lane n, vgpr r) = D[r + 8*(n>=16), n%16]
        #pragma unroll
        for (int r = 0; r < 8; ++r)
            bounce[wave][lane * 8 + r] = c[r];
        __syncthreads();

        if (lane < 16) {
            // diagonal pick: value for negative m=lane
            const int idx = (lane < 8) ? (lane * 8 + lane)
                                       : ((lane + 8) * 8 + (lane - 8));
            const float dval = bounce[wave][idx];
            out_dist[(size_t)b * KNEG + t * 16 + lane] = dval;
            minv = fminf(minv, dval);
        }
        __syncthreads();   // bounce reused next tile
    }

    // ---- min over K: lanes 0-15 hold per-slot minima ----
    if (lane >= 16) minv = INFINITY;
    #pragma unroll
    for (int mask = 8; mask; mask >>= 1)
        minv = fminf(minv, __shfl_xor(minv, mask, 32));
    if (lane == 0) out_min[b] = minv;
}

extern "C" void kernel_launch(void* const* d_in, const int* in_sizes, int n_in,
                              void* d_out, int out_size, void* d_ws, size_t ws_size,
                              hipStream_t stream) {
    const int*   pairs     = (const int*)d_in[0];
    const int*   pos_art   = (const int*)d_in[1];
    const int*   pos_alb   = (const int*)d_in[2];
    const int*   neg_items = (const int*)d_in[3];
    const int*   neg_art   = (const int*)d_in[4];
    const int*   neg_alb   = (const int*)d_in[5];
    // d_in[6] titles, d_in[7] titles_len: unused by the reference outputs
    const float* Wu   = (const float*)d_in[8];
    const float* Wi   = (const float*)d_in[9];
    const float* Wart = (const float*)d_in[10];
    const float* Walb = (const float*)d_in[11];

    const int B = in_sizes[1];            // 4096
    float* out_pos  = (float*)d_out;                    // [0, B)
    float* out_dist = out_pos + B;                      // [B, B + B*K)
    float* out_min  = out_dist + (size_t)B * KNEG;      // [B + B*K, B + B*K + B)

    const int blocks = (B + 7) / 8;       // 8 users (waves) per 256-thread block
    cml_dist_kernel<<<blocks, 256, 0, stream>>>(
        pairs, pos_art, pos_alb, neg_items, neg_art, neg_alb,
        Wu, Wi, Wart, Walb, out_pos, out_dist, out_min, B);
}